// DampedIMEX2Layer_22368189677866
// MI455X (gfx1250) — compile-verified
//
#include <hip/hip_runtime.h>
#include <hip/hip_bf16.h>

// Problem constants
#define SL   16384   // sequence length L
#define HD   256     // hidden dim H
#define PD   512     // state channels P
#define NC   128     // number of scan chunks
#define CHUNK (SL / NC)          // 128 timesteps per chunk
#define LP   ((size_t)SL * PD)   // plane size (floats)
#define PH   ((size_t)PD * HD)   // weight plane size (floats)

typedef float v2f __attribute__((ext_vector_type(2)));
typedef float v8f __attribute__((ext_vector_type(8)));

// ---------------------------------------------------------------------------
// Kernel 0: deinterleave complex weights once so GEMM fragments are straight
// aligned b64 loads (no per-iteration VALU packing). Cim is pre-negated.
// ---------------------------------------------------------------------------
__global__ __launch_bounds__(256) void deinterleave_wts(const float* __restrict__ B2,
                                                        const float* __restrict__ C2,
                                                        float* __restrict__ Bre,
                                                        float* __restrict__ Bim,
                                                        float* __restrict__ Cre,
                                                        float* __restrict__ CimN) {
    const int i = blockIdx.x * blockDim.x + threadIdx.x;
    if (i >= (int)PH) return;
    float2 b = ((const float2*)B2)[i];
    float2 c = ((const float2*)C2)[i];
    Bre[i]  = b.x;
    Bim[i]  = b.y;
    Cre[i]  = c.x;
    CimN[i] = -c.y;
}

// ---------------------------------------------------------------------------
// Kernel 1: per-channel parameters.
// prm layout (floats): [0,P) dts | [P,5P) M | [5P,9P) MT = M^CHUNK
// ---------------------------------------------------------------------------
__global__ void precompute_params(const float* __restrict__ A_diag,
                                  const float* __restrict__ G_diag,
                                  const float* __restrict__ dt,
                                  float* __restrict__ prm) {
    int p = blockIdx.x * blockDim.x + threadIdx.x;
    if (p >= PD) return;
    float dts  = 1.0f / (1.0f + expf(-dt[p]));
    float G    = fmaxf(G_diag[p], 0.0f);
    float root = sqrtf(1.0f - dts * G);
    float den  = fmaxf(dts * dts, 1e-6f);
    float A_lo = (2.0f - dts * G - 2.0f * root) / den;
    float A_hi = (2.0f - dts * G + 2.0f * root) / den;
    float Ad   = A_diag[p];
    float A    = A_lo + fmaxf(Ad - A_lo, 0.0f) - fmaxf(Ad - A_hi, 0.0f);

    float m00 = 1.0f - dts * G;
    float m01 = -dts * A;
    float m10 = dts * (1.0f - dts * G);
    float m11 = 1.0f - dts * dts * A;

    prm[p]              = dts;
    prm[PD + 4 * p + 0] = m00;
    prm[PD + 4 * p + 1] = m01;
    prm[PD + 4 * p + 2] = m10;
    prm[PD + 4 * p + 3] = m11;

    // MT = M^CHUNK (2x2 repeated multiply; 128 iters, trivial)
    float t00 = 1.f, t01 = 0.f, t10 = 0.f, t11 = 1.f;
    #pragma unroll 4
    for (int i = 0; i < CHUNK; ++i) {
        float n00 = m00 * t00 + m01 * t10;
        float n01 = m00 * t01 + m01 * t11;
        float n10 = m10 * t00 + m11 * t10;
        float n11 = m10 * t01 + m11 * t11;
        t00 = n00; t01 = n01; t10 = n10; t11 = n11;
    }
    prm[5 * PD + 4 * p + 0] = t00;
    prm[5 * PD + 4 * p + 1] = t01;
    prm[5 * PD + 4 * p + 2] = t10;
    prm[5 * PD + 4 * p + 3] = t11;
}

// ---------------------------------------------------------------------------
// Kernel 2: Bu = u @ B_c.T via fp32 WMMA 16x16x4.
// Each wave computes a 32x16 tile (two M-tiles), four accumulators
// (Re/Im x 2 row-tiles). B fragments reused across both row tiles; all
// fragment loads are aligned b64 with no packing.
// ---------------------------------------------------------------------------
__global__ __launch_bounds__(256) void gemm_bu(const float* __restrict__ u,
                                               const float* __restrict__ Bre,
                                               const float* __restrict__ Bim,
                                               float* __restrict__ buRe,
                                               float* __restrict__ buIm) {
    const int lane = threadIdx.x & 31;
    const int wave = threadIdx.x >> 5;
    const int NTN  = PD / 16;                       // 32 N-tiles
    const int job  = blockIdx.x * 8 + wave;         // (SL/32)*(PD/16) jobs
    const int tms  = job / NTN;                     // 32-row supertile
    const int tn   = job - tms * NTN;

    const int half = lane >> 4;        // 0: K,K+1   1: K+2,K+3
    const int lr   = lane & 15;
    const int row0 = tms * 32 + lr;
    const int row1 = row0 + 16;
    const int colB = tn * 16 + lr;     // P column this lane supplies for B

    const float* uR0 = u + (size_t)row0 * HD;
    const float* uR1 = u + (size_t)row1 * HD;
    const float* Bpr = Bre + (size_t)colB * HD;
    const float* Bpi = Bim + (size_t)colB * HD;

    v8f aR0 = {0.f,0.f,0.f,0.f,0.f,0.f,0.f,0.f};
    v8f aI0 = aR0, aR1 = aR0, aI1 = aR0;

    #pragma unroll 4
    for (int k = 0; k < HD; k += 4) {
        const int kk = k + 2 * half;
        v2f a0 = *(const v2f*)(uR0 + kk);
        v2f a1 = *(const v2f*)(uR1 + kk);
        v2f br = *(const v2f*)(Bpr + kk);
        v2f bi = *(const v2f*)(Bpi + kk);
        aR0 = __builtin_amdgcn_wmma_f32_16x16x4_f32(false, a0, false, br, (short)0, aR0, false, false);
        aI0 = __builtin_amdgcn_wmma_f32_16x16x4_f32(false, a0, false, bi, (short)0, aI0, false, false);
        aR1 = __builtin_amdgcn_wmma_f32_16x16x4_f32(false, a1, false, br, (short)0, aR1, false, false);
        aI1 = __builtin_amdgcn_wmma_f32_16x16x4_f32(false, a1, false, bi, (short)0, aI1, false, false);
    }

    const int col = tn * 16 + lr;
    #pragma unroll
    for (int r = 0; r < 8; ++r) {
        const int ra = tms * 32 + r + 8 * half;
        const int rb = ra + 16;
        buRe[(size_t)ra * PD + col] = aR0[r];
        buIm[(size_t)ra * PD + col] = aI0[r];
        buRe[(size_t)rb * PD + col] = aR1[r];
        buIm[(size_t)rb * PD + col] = aI1[r];
    }
}

// ---------------------------------------------------------------------------
// Kernel 3: chunk-local scan reduce. One block per chunk, one thread per
// channel; reads Bu coalesced across P, produces end-of-chunk state summary.
// ---------------------------------------------------------------------------
__global__ __launch_bounds__(PD) void scan_local(const float* __restrict__ buRe,
                                                 const float* __restrict__ buIm,
                                                 const float* __restrict__ prm,
                                                 float* __restrict__ summ) {
    const int p = threadIdx.x;
    const int c = blockIdx.x;
    const float dts = prm[p];
    const float d2  = dts * dts;
    const float m00 = prm[PD + 4 * p + 0];
    const float m01 = prm[PD + 4 * p + 1];
    const float m10 = prm[PD + 4 * p + 2];
    const float m11 = prm[PD + 4 * p + 3];

    float x1r = 0.f, x1i = 0.f, x2r = 0.f, x2i = 0.f;
    size_t base = (size_t)c * CHUNK * PD + p;
    #pragma unroll 4
    for (int i = 0; i < CHUNK; ++i) {
        const float br = buRe[base];
        const float bi = buIm[base];
        const float n1r = m00 * x1r + m01 * x2r + dts * br;
        const float n1i = m00 * x1i + m01 * x2i + dts * bi;
        const float n2r = m10 * x1r + m11 * x2r + d2 * br;
        const float n2i = m10 * x1i + m11 * x2i + d2 * bi;
        x1r = n1r; x1i = n1i; x2r = n2r; x2i = n2i;
        base += PD;
    }
    float4 s = {x1r, x1i, x2r, x2i};
    ((float4*)summ)[(size_t)c * PD + p] = s;
}

// ---------------------------------------------------------------------------
// Kernel 4: sequential prefix over NC chunk summaries using MT = M^CHUNK.
// carry[c] = state entering chunk c.
// ---------------------------------------------------------------------------
__global__ __launch_bounds__(PD) void scan_prefix(const float* __restrict__ prm,
                                                  const float* __restrict__ summ,
                                                  float* __restrict__ carry) {
    const int p = threadIdx.x;
    const float t00 = prm[5 * PD + 4 * p + 0];
    const float t01 = prm[5 * PD + 4 * p + 1];
    const float t10 = prm[5 * PD + 4 * p + 2];
    const float t11 = prm[5 * PD + 4 * p + 3];

    float c1r = 0.f, c1i = 0.f, c2r = 0.f, c2i = 0.f;
    for (int c = 0; c < NC; ++c) {
        float4 cv = {c1r, c1i, c2r, c2i};
        ((float4*)carry)[(size_t)c * PD + p] = cv;
        float4 s = ((const float4*)summ)[(size_t)c * PD + p];
        const float n1r = t00 * c1r + t01 * c2r + s.x;
        const float n1i = t00 * c1i + t01 * c2i + s.y;
        const float n2r = t10 * c1r + t11 * c2r + s.z;
        const float n2i = t10 * c1i + t11 * c2i + s.w;
        c1r = n1r; c1i = n1i; c2r = n2r; c2i = n2i;
    }
}

// ---------------------------------------------------------------------------
// Kernel 5: apply pass — redo the local scan starting from carry, writing
// ys = x[1] (complex) in place over the Bu planes.
// ---------------------------------------------------------------------------
__global__ __launch_bounds__(PD) void scan_apply(float* __restrict__ buRe,
                                                 float* __restrict__ buIm,
                                                 const float* __restrict__ prm,
                                                 const float* __restrict__ carry) {
    const int p = threadIdx.x;
    const int c = blockIdx.x;
    const float dts = prm[p];
    const float d2  = dts * dts;
    const float m00 = prm[PD + 4 * p + 0];
    const float m01 = prm[PD + 4 * p + 1];
    const float m10 = prm[PD + 4 * p + 2];
    const float m11 = prm[PD + 4 * p + 3];

    float4 cv = ((const float4*)carry)[(size_t)c * PD + p];
    float x1r = cv.x, x1i = cv.y, x2r = cv.z, x2i = cv.w;

    size_t base = (size_t)c * CHUNK * PD + p;
    #pragma unroll 4
    for (int i = 0; i < CHUNK; ++i) {
        const float br = buRe[base];
        const float bi = buIm[base];
        const float n1r = m00 * x1r + m01 * x2r + dts * br;
        const float n1i = m00 * x1i + m01 * x2i + dts * bi;
        const float n2r = m10 * x1r + m11 * x2r + d2 * br;
        const float n2i = m10 * x1i + m11 * x2i + d2 * bi;
        x1r = n1r; x1i = n1i; x2r = n2r; x2i = n2i;
        buRe[base] = x2r;          // ys = second state component
        buIm[base] = x2i;
        base += PD;
    }
}

// ---------------------------------------------------------------------------
// Kernel 6: out = Re(ys @ C_c.T) + u * D, via fp32 WMMA. K = P = 512.
// Cre / pre-negated Cim planes -> subtraction folds into accumulation.
// Each wave: 32x16 tile, two accumulators, C fragments shared.
// ---------------------------------------------------------------------------
__global__ __launch_bounds__(256) void gemm_out(const float* __restrict__ ysRe,
                                                const float* __restrict__ ysIm,
                                                const float* __restrict__ Cre,
                                                const float* __restrict__ CimN,
                                                const float* __restrict__ u,
                                                const float* __restrict__ D,
                                                float* __restrict__ out) {
    const int lane = threadIdx.x & 31;
    const int wave = threadIdx.x >> 5;
    const int NTN  = HD / 16;                       // 16 N-tiles
    const int job  = blockIdx.x * 8 + wave;         // (SL/32)*(HD/16) jobs
    const int tms  = job / NTN;
    const int tn   = job - tms * NTN;

    const int half = lane >> 4;
    const int lr   = lane & 15;
    const int row0 = tms * 32 + lr;
    const int row1 = row0 + 16;
    const int colB = tn * 16 + lr;                  // H column

    const float* yr0 = ysRe + (size_t)row0 * PD;
    const float* yi0 = ysIm + (size_t)row0 * PD;
    const float* yr1 = ysRe + (size_t)row1 * PD;
    const float* yi1 = ysIm + (size_t)row1 * PD;
    const float* Cr  = Cre  + (size_t)colB * PD;
    const float* Ci  = CimN + (size_t)colB * PD;

    v8f acc0 = {0.f,0.f,0.f,0.f,0.f,0.f,0.f,0.f};
    v8f acc1 = acc0;

    #pragma unroll 4
    for (int k = 0; k < PD; k += 4) {
        const int kk = k + 2 * half;
        v2f cr  = *(const v2f*)(Cr  + kk);
        v2f ci  = *(const v2f*)(Ci  + kk);
        v2f ar0 = *(const v2f*)(yr0 + kk);
        v2f ai0 = *(const v2f*)(yi0 + kk);
        v2f ar1 = *(const v2f*)(yr1 + kk);
        v2f ai1 = *(const v2f*)(yi1 + kk);
        acc0 = __builtin_amdgcn_wmma_f32_16x16x4_f32(false, ar0, false, cr, (short)0, acc0, false, false);
        acc1 = __builtin_amdgcn_wmma_f32_16x16x4_f32(false, ar1, false, cr, (short)0, acc1, false, false);
        acc0 = __builtin_amdgcn_wmma_f32_16x16x4_f32(false, ai0, false, ci, (short)0, acc0, false, false);
        acc1 = __builtin_amdgcn_wmma_f32_16x16x4_f32(false, ai1, false, ci, (short)0, acc1, false, false);
    }

    const int col = tn * 16 + lr;
    const float Dv = D[col];
    #pragma unroll
    for (int r = 0; r < 8; ++r) {
        const int ra = tms * 32 + r + 8 * half;
        const int rb = ra + 16;
        out[(size_t)ra * HD + col] = acc0[r] + u[(size_t)ra * HD + col] * Dv;
        out[(size_t)rb * HD + col] = acc1[r] + u[(size_t)rb * HD + col] * Dv;
    }
}

// ---------------------------------------------------------------------------
extern "C" void kernel_launch(void* const* d_in, const int* in_sizes, int n_in,
                              void* d_out, int out_size, void* d_ws, size_t ws_size,
                              hipStream_t stream) {
    const float* u      = (const float*)d_in[0];   // (L,H)
    const float* A_diag = (const float*)d_in[1];   // (P,)
    const float* G_diag = (const float*)d_in[2];   // (P,)
    const float* dt     = (const float*)d_in[3];   // (P,)
    const float* B      = (const float*)d_in[4];   // (P,H,2)
    const float* Cm     = (const float*)d_in[5];   // (H,P,2)
    const float* D      = (const float*)d_in[6];   // (H,)
    float* out          = (float*)d_out;           // (L,H)

    // Workspace layout (floats)
    float* ws    = (float*)d_ws;
    float* buRe  = ws;                              // LP (becomes ys_re)
    float* buIm  = ws + LP;                         // LP (becomes ys_im)
    float* prm   = ws + 2 * LP;                     // 9*P
    float* summ  = prm + 9 * PD;                    // NC*P*4
    float* carry = summ + (size_t)NC * PD * 4;      // NC*P*4
    float* Bre   = carry + (size_t)NC * PD * 4;     // PH
    float* Bim   = Bre + PH;                        // PH
    float* Cre   = Bim + PH;                        // PH
    float* CimN  = Cre + PH;                        // PH

    // 0. deinterleave complex weights (Cim pre-negated)
    deinterleave_wts<<<((int)PH + 255) / 256, 256, 0, stream>>>(B, Cm, Bre, Bim, Cre, CimN);

    // 1. per-channel params + chunk matrix power
    precompute_params<<<(PD + 255) / 256, 256, 0, stream>>>(A_diag, G_diag, dt, prm);

    // 2. Bu = u @ B_c.T   (WMMA f32)
    {
        const int jobs = (SL / 32) * (PD / 16);    // 16384
        gemm_bu<<<jobs / 8, 256, 0, stream>>>(u, Bre, Bim, buRe, buIm);
    }

    // 3-5. chunked linear-recurrence scan
    scan_local <<<NC, PD, 0, stream>>>(buRe, buIm, prm, summ);
    scan_prefix<<<1, PD, 0, stream>>>(prm, summ, carry);
    scan_apply <<<NC, PD, 0, stream>>>(buRe, buIm, prm, carry);

    // 6. out = Re(ys @ C_c.T) + u * D   (WMMA f32)
    {
        const int jobs = (SL / 32) * (HD / 16);    // 8192
        gemm_out<<<jobs / 8, 256, 0, stream>>>(buRe, buIm, Cre, CimN, u, D, out);
    }
}